// BatchBlur_86517821213120
// MI455X (gfx1250) — compile-verified
//
#include <hip/hip_runtime.h>

// BatchBlur (per-sample 15x15 depthwise conv, reflect pad) for gfx1250.
// Stage 1: horizontal conv of each padded row with all 15 kernel rows as a
//          16x16x16 f32 GEMM built from 4x V_WMMA_F32_16X16X4_F32, with the
//          global row load software-pipelined one iteration ahead.
//          Results are scattered into per-output-row buckets in LDS.
// Stage 2: per output, 4x ds_load_b128 + pairwise sum of 16 floats
//          (the dy==15 slot is exactly 0 because B column 15 is zero).

typedef float v2f __attribute__((ext_vector_type(2)));
typedef float v4f __attribute__((ext_vector_type(4)));
typedef float v8f __attribute__((ext_vector_type(8)));

#define LK      15          // blur kernel size
#define PADR    7           // L/2
#define HH      512
#define WW      512
#define CH      3
#define NB      32
#define TS      32          // output rows per tile
#define GR      (TS + LK - 1)   // 46 G-rows per tile
#define WAVES   4

__device__ __forceinline__ int reflect512(int ip) {
    // padded index -> source index, jnp.pad(..., mode="reflect") semantics
    int i = ip - PADR;
    i = (i < 0) ? -i : i;
    i = (i >= HH) ? (2 * HH - 2 - i) : i;
    return i;
}

__global__ __launch_bounds__(32 * WAVES)
void batchblur_wmma(const float* __restrict__ x,
                    const float* __restrict__ kern,
                    float* __restrict__ out)
{
    const int tid  = threadIdx.x;
    const int lane = tid & 31;
    const int wv   = tid >> 5;           // wave id 0..3
    const int half = (lane >> 4) & 1;    // lane half
    const int m16  = lane & 15;

    const int tile_x = blockIdx.x;       // WW/16 = 32 column tiles
    const int tile_y = blockIdx.y;       // HH/TS = 16 row tiles
    const int bc     = blockIdx.z;       // 96 (b,c) planes
    const int b      = bc / CH;

    const float* plane  = x   + (size_t)bc * (HH * WW);
    float*       oplane = out + (size_t)bc * (HH * WW);
    const float* kb     = kern + (size_t)b * (LK * LK);

    // Per-output-row buckets: gbuf2[yy][m][dy], 32*16*16 floats = 32 KB.
    __shared__ __align__(16) float gbuf2[TS * 16 * 16];

    // ---- B matrix chunks (per wave): B[k][n] = K[n][k], zero-padded 16x16.
    // B 4x16 layout: VGPR v, lane l -> k = 2*(l/16)+v (+4c), n = l%16.
    // Column n==15 and row k==15 are zero -> D[:,15] == 0 exactly.
    v2f Bc[4];
    #pragma unroll
    for (int c = 0; c < 4; ++c) {
        #pragma unroll
        for (int v = 0; v < 2; ++v) {
            const int k = 4 * c + 2 * half + v;      // tap dx
            float kv = 0.0f;
            if (k < LK && m16 < LK) kv = kb[m16 * LK + k];
            Bc[c][v] = kv;
        }
    }

    const int x0 = tile_x * 16;   // output col base == padded col base
    const int y0 = tile_y * TS;   // output row base; padded rows y0..y0+GR-1
    const int colg = reflect512(x0 + lane);   // staging column for this lane

    // ---- Stage 1 with one-deep software pipeline on the row load.
    float rowval = plane[(size_t)reflect512(y0 + wv) * WW + colg];
    for (int r = wv; r < GR; r += WAVES) {
        const int rn = (r + WAVES < GR) ? (r + WAVES) : (GR - 1);
        const float nextval = plane[(size_t)reflect512(y0 + rn) * WW + colg];

        v8f d = {0.f, 0.f, 0.f, 0.f, 0.f, 0.f, 0.f, 0.f};
        #pragma unroll
        for (int c = 0; c < 4; ++c) {
            // A 16x4 layout: VGPR v, lane l -> m = l%16, k = 2*(l/16)+v (+4c)
            // A[m][k] = paddedrow[x0 + m + (4c + k_local)]  (Toeplitz)
            const int base = m16 + 4 * c + 2 * half;   // <= 30
            v2f a;
            a[0] = __shfl(rowval, base,     32);
            a[1] = __shfl(rowval, base + 1, 32);
            d = __builtin_amdgcn_wmma_f32_16x16x4_f32(
                    /*neg_a=*/false, a, /*neg_b=*/false, Bc[c],
                    /*c_mod=*/(short)0, d, /*reuse_a=*/false, /*reuse_b=*/false);
        }

        // C/D layout: VGPR j, lanes0-15 -> M=j, lanes16-31 -> M=j+8; N=l%16.
        // Element (m = output col, dy = m16) belongs to output row yy = r-dy.
        // Scatter into gbuf2[yy][m][dy]; dy==15 lane stores exact 0 (pad slot).
        const int yy = r - m16;
        if (yy >= 0 && yy < TS) {
            float* dst = gbuf2 + yy * 256 + half * (8 * 16) + m16;
            #pragma unroll
            for (int j = 0; j < 8; ++j)
                dst[j * 16] = d[j];
        }
        rowval = nextval;
    }
    __syncthreads();

    // ---- Stage 2: out(y0+yy, x0+m) = sum of 16-float bucket gbuf2[yy][m][*]
    const int lane_row = wv * 2 + half;   // 0..7
    #pragma unroll
    for (int t = 0; t < TS / (2 * WAVES); ++t) {   // 4 rows per (wave,half)
        const int yy = lane_row + t * 2 * WAVES;
        const v4f* p = (const v4f*)(gbuf2 + (yy * 16 + m16) * 16);
        const v4f s0 = p[0], s1 = p[1], s2 = p[2], s3 = p[3];
        const v4f s = (s0 + s1) + (s2 + s3);
        const float acc = (s.x + s.y) + (s.z + s.w);
        oplane[(size_t)(y0 + yy) * WW + (x0 + m16)] = acc;
    }
}

extern "C" void kernel_launch(void* const* d_in, const int* in_sizes, int n_in,
                              void* d_out, int out_size, void* d_ws, size_t ws_size,
                              hipStream_t stream) {
    (void)in_sizes; (void)n_in; (void)out_size; (void)d_ws; (void)ws_size;
    const float* x    = (const float*)d_in[0];   // (32,3,512,512) f32
    const float* kern = (const float*)d_in[1];   // (32,15,15) f32
    float*       out  = (float*)d_out;           // (32,3,512,512) f32

    dim3 grid(WW / 16, HH / TS, NB * CH);
    dim3 block(32 * WAVES);
    batchblur_wmma<<<grid, block, 0, stream>>>(x, kern, out);
}